// Block_16595753632223
// MI455X (gfx1250) — compile-verified
//
#include <hip/hip_runtime.h>
#include <cstddef>
#include <cstdint>

// ---------------------------------------------------------------------------
// MI455X / gfx1250 transformer block forward, fp32 end-to-end.
// - All GEMMs (QKV, O-proj, FC1, FC2) and both attention GEMMs run on
//   V_WMMA_F32_16X16X4_F32 (wave32 matrix pipe).
// - GEMM tiles are staged with GLOBAL_LOAD_ASYNC_TO_LDS_B128 (ASYNCcnt) and
//   double-buffered: issue next slab's async copies, compute current slab,
//   s_wait_asynccnt 0 + one barrier per K-step.
// ---------------------------------------------------------------------------

typedef float v2f __attribute__((ext_vector_type(2)));
typedef float v8f __attribute__((ext_vector_type(8)));

#define WMMA_F32(a, b, c) \
  __builtin_amdgcn_wmma_f32_16x16x4_f32(false, (a), false, (b), (short)0, (c), false, false)

constexpr int kB = 2;
constexpr int kT = 2048;
constexpr int kC = 1024;
constexpr int kH = 16;
constexpr int kD = 64;          // kC / kH
constexpr int kBT = kB * kT;    // 4096 rows

// --- CDNA5 async L2->LDS copy (per-lane 16B), tracked by ASYNCcnt ----------
// LDS generic address low 32 bits == LDS byte offset (ISA aperture rule).
__device__ __forceinline__ void async_ld_b128(float* lds_ptr, const float* gptr) {
  const uint32_t loff = (uint32_t)(uintptr_t)lds_ptr;
  asm volatile("global_load_async_to_lds_b128 %0, %1, off"
               :: "v"(loff), "v"(gptr) : "memory");
}
__device__ __forceinline__ void wait_async0() {
  asm volatile("s_wait_asynccnt 0x0" ::: "memory");
}

// ===========================================================================
// LayerNorm: one 256-thread block per row of C=1024 (float4 per thread).
// ===========================================================================
__global__ __launch_bounds__(256)
void ln_kernel(const float* __restrict__ x, const float* __restrict__ g,
               const float* __restrict__ b, float* __restrict__ out)
{
  __shared__ float red0[8];
  __shared__ float red1[8];
  const int row = blockIdx.x;
  const int t = threadIdx.x;
  const float* xr = x + (size_t)row * kC;
  float4 v = *(const float4*)(xr + t * 4);
  float s  = v.x + v.y + v.z + v.w;
  float ss = v.x * v.x + v.y * v.y + v.z * v.z + v.w * v.w;
  #pragma unroll
  for (int off = 16; off; off >>= 1) {
    s  += __shfl_xor(s,  off, 32);
    ss += __shfl_xor(ss, off, 32);
  }
  if ((t & 31) == 0) { red0[t >> 5] = s; red1[t >> 5] = ss; }
  __syncthreads();
  s  = red0[t & 7];
  ss = red1[t & 7];
  #pragma unroll
  for (int off = 4; off; off >>= 1) {
    s  += __shfl_xor(s,  off, 32);
    ss += __shfl_xor(ss, off, 32);
  }
  const float mu  = s * (1.0f / kC);
  const float var = ss * (1.0f / kC) - mu * mu;
  const float r   = rsqrtf(var + 1e-5f);
  const float4 gg = *(const float4*)(g + t * 4);
  const float4 bb = *(const float4*)(b + t * 4);
  float4 o;
  o.x = (v.x - mu) * r * gg.x + bb.x;
  o.y = (v.y - mu) * r * gg.y + bb.y;
  o.z = (v.z - mu) * r * gg.z + bb.z;
  o.w = (v.w - mu) * r * gg.w + bb.w;
  *(float4*)(out + (size_t)row * kC + t * 4) = o;
}

// ===========================================================================
// WMMA fp32 GEMM: C[M,N] = epilogue(A[M,K] @ W[K,N] + bias [+ res]).
// Block = 256 threads (8 waves), tile 128(M) x 64(N), BK=16, double-buffered
// async-to-LDS staging. Wave grid 4x2; each wave -> 32x32 patch = 2x2 WMMA
// tiles. EPI: 0 = bias, 1 = bias + exact GELU, 2 = bias + residual.
// ===========================================================================
constexpr int GBM = 128, GBN = 64, GBK = 16;
constexpr int APAD = 20;   // A row stride (floats): 16B-aligned, bank-clean b64 reads
constexpr int BPAD = 72;   // B row stride (floats): hi half-wave lands +16 banks

template <int EPI>
__global__ __launch_bounds__(256)
void gemm_wmma(const float* __restrict__ A, const float* __restrict__ W,
               const float* __restrict__ bias, const float* __restrict__ res,
               float* __restrict__ C, int M, int N, int K)
{
  __shared__ __align__(16) float As[2][GBM * APAD];   // row-major [128][20]
  __shared__ __align__(16) float Bs[2][GBK * BPAD];   // row-major [k][n] [16][72]
  const int t    = threadIdx.x;
  const int m0   = blockIdx.y * GBM;
  const int n0   = blockIdx.x * GBN;
  const int w    = t >> 5;
  const int lane = t & 31;
  const int wm   = w >> 1;       // 0..3
  const int wn   = w & 1;        // 0..1
  const bool lo  = lane < 16;
  const int  lr  = lane & 15;
  const int  ko0 = lo ? 0 : 2;   // 16x16x4 frag layout: hi half-wave holds K2,K3

  // Per-thread staging coordinates (fixed across K-steps).
  const int ar0 = t >> 2;                 // A chunk 0: rows 0..63
  const int ac0 = (t & 3) << 2;
  const int ar1 = (t + 256) >> 2;         // A chunk 1: rows 64..127
  const int brow = t >> 4;                // B: k row 0..15
  const int bc4  = (t & 15) << 2;         // B: n col {0,4,...,60}

  v8f acc[2][2];
  #pragma unroll
  for (int i = 0; i < 2; ++i)
    #pragma unroll
    for (int j = 0; j < 2; ++j)
      #pragma unroll
      for (int v = 0; v < 8; ++v) acc[i][j][v] = 0.0f;

  // Issue async staging of one K-slab into buffer `buf`.
  auto stage = [&](int buf, int k0) {
    async_ld_b128(&As[buf][ar0 * APAD + ac0], A + (size_t)(m0 + ar0) * K + k0 + ac0);
    async_ld_b128(&As[buf][ar1 * APAD + ac0], A + (size_t)(m0 + ar1) * K + k0 + ac0);
    async_ld_b128(&Bs[buf][brow * BPAD + bc4], W + (size_t)(k0 + brow) * N + n0 + bc4);
  };

  const int steps = K / GBK;
  stage(0, 0);

  for (int i = 0; i < steps; ++i) {
    wait_async0();        // own async copies (into buffer i&1) complete
    __syncthreads();      // -> every wave's copies complete; prev compute done
    if (i + 1 < steps) stage((i + 1) & 1, (i + 1) * GBK);

    const float* as = As[i & 1];
    const float* bs = Bs[i & 1];
    #pragma unroll
    for (int kc = 0; kc < 4; ++kc) {
      const int ko = kc * 4 + ko0;
      const v2f a0 = *(const v2f*)&as[(wm * 32 +  0 + lr) * APAD + ko];
      const v2f a1 = *(const v2f*)&as[(wm * 32 + 16 + lr) * APAD + ko];
      v2f b0, b1;
      b0.x = bs[(ko + 0) * BPAD + wn * 32 +  0 + lr];
      b0.y = bs[(ko + 1) * BPAD + wn * 32 +  0 + lr];
      b1.x = bs[(ko + 0) * BPAD + wn * 32 + 16 + lr];
      b1.y = bs[(ko + 1) * BPAD + wn * 32 + 16 + lr];
      acc[0][0] = WMMA_F32(a0, b0, acc[0][0]);
      acc[0][1] = WMMA_F32(a0, b1, acc[0][1]);
      acc[1][0] = WMMA_F32(a1, b0, acc[1][0]);
      acc[1][1] = WMMA_F32(a1, b1, acc[1][1]);
    }
  }

  // Epilogue. D-layout: lanes 0-15 -> rows 0-7 (VGPR v), col = lane;
  //                     lanes 16-31 -> rows 8-15, col = lane-16.
  #pragma unroll
  for (int tm = 0; tm < 2; ++tm) {
    #pragma unroll
    for (int tn = 0; tn < 2; ++tn) {
      const int col   = n0 + wn * 32 + tn * 16 + lr;
      const float bv  = bias[col];
      const int rbase = m0 + wm * 32 + tm * 16 + (lo ? 0 : 8);
      #pragma unroll
      for (int v = 0; v < 8; ++v) {
        const int row = rbase + v;
        float val = acc[tm][tn][v] + bv;
        if (EPI == 1) val = 0.5f * val * (1.0f + erff(val * 0.70710678118654752f));
        if (EPI == 2) val += res[(size_t)row * N + col];
        C[(size_t)row * N + col] = val;
      }
    }
  }
}

// ===========================================================================
// Causal flash attention (fp32 WMMA), transposed orientation.
// One wave per 16-query tile. S^T = K*Q^T puts queries in lanes, keys in
// VGPRs -> softmax stats and O rescale are per-lane; P^T registers feed the
// B-operand of O^T += V^T * P^T after 8 xor-16 shuffles.
// qkv layout: [B, T, 3C]; q at h*64, k at C + h*64, v at 2C + h*64.
// ===========================================================================
__global__ __launch_bounds__(256)
void attn_fwd(const float* __restrict__ qkv, float* __restrict__ y)
{
  const int w    = threadIdx.x >> 5;
  const int lane = threadIdx.x & 31;
  const bool lo  = lane < 16;
  const int  lr  = lane & 15;
  const int  qt  = blockIdx.x * 8 + w;     // 0..127 query tile
  const int  h   = blockIdx.y;
  const int  bb  = blockIdx.z;
  const int  qg  = qt * 16 + lr;           // this lane's query row
  const int  RS  = 3 * kC;                 // qkv row stride (floats)
  const float* base = qkv + (size_t)bb * kT * RS + h * kD;
  const int coff = lo ? 0 : 2;

  // Q^T B-frags: lane holds Q[q=lr][4dc + coff .. +1] (contiguous float2).
  v2f qf[16];
  const float* qrow = base + (size_t)qg * RS;
  #pragma unroll
  for (int dc = 0; dc < 16; ++dc)
    qf[dc] = *(const v2f*)(qrow + 4 * dc + coff);

  v8f o[4];                                 // O^T accum: 4 d-tiles of 16
  #pragma unroll
  for (int dt = 0; dt < 4; ++dt)
    #pragma unroll
    for (int v = 0; v < 8; ++v) o[dt][v] = 0.0f;

  float m = -1e30f, l = 0.0f;

  for (int kb = 0; kb <= qt; ++kb) {
    // ---- S^T tile = K * Q^T : A-frag from K rows (contiguous float2) ----
    v8f s;
    #pragma unroll
    for (int v = 0; v < 8; ++v) s[v] = 0.0f;
    const float* krow = base + kC + (size_t)(kb * 16 + lr) * RS;
    #pragma unroll
    for (int dc = 0; dc < 16; ++dc) {
      const v2f kf = *(const v2f*)(krow + 4 * dc + coff);
      s = WMMA_F32(kf, qf[dc], s);
    }

    // ---- scale + causal mask + online softmax (keys live in VGPRs) ----
    float p[8];
    float rmax = -1e30f;
    const int kbase = kb * 16 + (lo ? 0 : 8);
    #pragma unroll
    for (int v = 0; v < 8; ++v) {
      float sv = s[v] * 0.125f;             // 1/sqrt(64)
      sv = (kbase + v <= qg) ? sv : -1e30f;
      p[v] = sv;
      rmax = fmaxf(rmax, sv);
    }
    rmax = fmaxf(rmax, __shfl_xor(rmax, 16, 32));   // other 8 keys of this q
    const float mnew  = fmaxf(m, rmax);
    const float alpha = __expf(m - mnew);
    float rsum = 0.0f;
    #pragma unroll
    for (int v = 0; v < 8; ++v) { p[v] = __expf(p[v] - mnew); rsum += p[v]; }
    rsum += __shfl_xor(rsum, 16, 32);
    l = l * alpha + rsum;
    m = mnew;
    #pragma unroll
    for (int dt = 0; dt < 4; ++dt)
      #pragma unroll
      for (int v = 0; v < 8; ++v) o[dt][v] *= alpha;

    // ---- Build P^T B-frags: each half-wave needs 2 VGPRs from the other ----
    float ps[8];
    #pragma unroll
    for (int v = 0; v < 8; ++v) ps[v] = __shfl_xor(p[v], 16, 32);
    v2f bf[4];
    bf[0].x = lo ? p[0]  : ps[2]; bf[0].y = lo ? p[1]  : ps[3];
    bf[1].x = lo ? p[4]  : ps[6]; bf[1].y = lo ? p[5]  : ps[7];
    bf[2].x = lo ? ps[0] : p[2];  bf[2].y = lo ? ps[1] : p[3];
    bf[3].x = lo ? ps[4] : p[6];  bf[3].y = lo ? ps[5] : p[7];

    // ---- O^T += V^T * P^T ----
    const float* vbase = base + 2 * kC + (size_t)(kb * 16) * RS;
    #pragma unroll
    for (int dt = 0; dt < 4; ++dt) {
      const int d = dt * 16 + lr;
      #pragma unroll
      for (int kc = 0; kc < 4; ++kc) {
        const float* vp = vbase + (size_t)(4 * kc + coff) * RS + d;
        v2f vf; vf.x = vp[0]; vf.y = vp[RS];   // V[key][d], V[key+1][d]
        o[dt] = WMMA_F32(vf, bf[kc], o[dt]);
      }
    }
  }

  // ---- write y[b, q, h*64 + d] = O^T / l ----
  const float inv = 1.0f / l;
  float* yrow = y + (size_t)(bb * kT + qg) * kC + h * kD + (lo ? 0 : 8);
  #pragma unroll
  for (int dt = 0; dt < 4; ++dt)
    #pragma unroll
    for (int v = 0; v < 8; ++v)
      yrow[dt * 16 + v] = o[dt][v] * inv;
}

// ===========================================================================
// Host-side orchestration (all launches on `stream`, graph-capture safe).
// ===========================================================================
extern "C" void kernel_launch(void* const* d_in, const int* in_sizes, int n_in,
                              void* d_out, int out_size, void* d_ws, size_t ws_size,
                              hipStream_t stream) {
  (void)in_sizes; (void)n_in; (void)out_size; (void)ws_size;
  const float* x      = (const float*)d_in[0];
  const float* ln1_g  = (const float*)d_in[1];
  const float* ln1_b  = (const float*)d_in[2];
  const float* W_attn = (const float*)d_in[3];
  const float* b_attn = (const float*)d_in[4];
  const float* W_o    = (const float*)d_in[5];
  const float* b_o    = (const float*)d_in[6];
  const float* ln2_g  = (const float*)d_in[7];
  const float* ln2_b  = (const float*)d_in[8];
  const float* W_fc   = (const float*)d_in[9];
  const float* b_fc   = (const float*)d_in[10];
  const float* W_fc2  = (const float*)d_in[11];
  const float* b_fc2  = (const float*)d_in[12];
  float* out = (float*)d_out;

  // Workspace carve-up (floats): ln-out, qkv, attn-out, residual-1, fc-hidden.
  float* ws   = (float*)d_ws;
  float* h_ln = ws;                              // BT*C    (reused for ln2)
  float* qkv  = h_ln + (size_t)kBT * kC;         // BT*3C
  float* yatt = qkv  + (size_t)kBT * 3 * kC;     // BT*C
  float* x1   = yatt + (size_t)kBT * kC;         // BT*C
  float* hfc  = x1   + (size_t)kBT * kC;         // BT*4C

  const dim3 blk(256);

  // 1) h = LN1(x)
  ln_kernel<<<dim3(kBT), blk, 0, stream>>>(x, ln1_g, ln1_b, h_ln);
  // 2) qkv = h @ W_attn + b_attn                    [4096 x 3072, K=1024]
  gemm_wmma<0><<<dim3(3 * kC / GBN, kBT / GBM), blk, 0, stream>>>(
      h_ln, W_attn, b_attn, nullptr, qkv, kBT, 3 * kC, kC);
  // 3) y = causal-softmax attention(qkv)
  attn_fwd<<<dim3(kT / 16 / 8, kH, kB), blk, 0, stream>>>(qkv, yatt);
  // 4) x1 = x + y @ W_o + b_o                       [4096 x 1024, K=1024]
  gemm_wmma<2><<<dim3(kC / GBN, kBT / GBM), blk, 0, stream>>>(
      yatt, W_o, b_o, x, x1, kBT, kC, kC);
  // 5) h = LN2(x1)
  ln_kernel<<<dim3(kBT), blk, 0, stream>>>(x1, ln2_g, ln2_b, h_ln);
  // 6) hfc = gelu(h @ W_fc + b_fc)                  [4096 x 4096, K=1024]
  gemm_wmma<1><<<dim3(4 * kC / GBN, kBT / GBM), blk, 0, stream>>>(
      h_ln, W_fc, b_fc, nullptr, hfc, kBT, 4 * kC, kC);
  // 7) out = x1 + hfc @ W_fc2 + b_fc2               [4096 x 1024, K=4096]
  gemm_wmma<2><<<dim3(kC / GBN, kBT / GBM), blk, 0, stream>>>(
      hfc, W_fc2, b_fc2, x1, out, kBT, kC, 4 * kC);
}